// gwDistance_76201309766055
// MI455X (gfx1250) — compile-verified
//
#include <hip/hip_runtime.h>
#include <math.h>
#include <stdint.h>

#define B_ 8
#define N_ 3000
#define M_ 3000
#define KNN 8
#define CAP 40          // 8 mask slots + up to 32 extras (Poisson(8) tail ~1e-11)
#define EPSV 0.1f
#define INV_EPS 10.0f
#define THRESH 0.1f
#define MAX_ITER 100

typedef __attribute__((ext_vector_type(2))) float v2f;
typedef __attribute__((ext_vector_type(8))) float v8f;

#if defined(__has_builtin)
#if __has_builtin(__builtin_amdgcn_wmma_f32_16x16x4_f32)
#define HAVE_WMMA4 1
#endif
#if __has_builtin(__builtin_amdgcn_global_load_async_to_lds_b64) && \
    __has_builtin(__builtin_amdgcn_s_wait_asynccnt)
#define HAVE_ASYNC 1
#endif
#endif

#ifdef HAVE_ASYNC
typedef int v2i __attribute__((vector_size(8)));
typedef __attribute__((address_space(3))) v2i lds_v2i;
__device__ __forceinline__ void async_g2l_b64(const void* g, const void* l) {
    __builtin_amdgcn_global_load_async_to_lds_b64(
        (v2i*)g, (lds_v2i*)(unsigned int)(uintptr_t)l, 0, 0);
}
#endif

// ---------------------------------------------------------------------------
// K1: row sums of squares via V_WMMA_F32_16X16X4_F32 (B = ones -> row sums).
// One wave per 16 rows. 16x64 tile staged in LDS (stride 68 avoids conflicts).
// Full chunks staged with GLOBAL_LOAD_ASYNC_TO_LDS_B64 (ASYNCcnt path).
// ---------------------------------------------------------------------------
__global__ __launch_bounds__(32) void k_rownorm(const float* __restrict__ C,
                                                float* __restrict__ rinv) {
    constexpr int BPB = (N_ + 15) / 16;  // 188 blocks per batch
    const int b = blockIdx.x / BPB;
    const int row0 = (blockIdx.x % BPB) * 16;
    const float* Cb = C + (size_t)b * N_ * M_;
    const int lane = threadIdx.x;

    __shared__ float tile[16 * 68];
    __shared__ float red[16];

#ifdef HAVE_WMMA4
    v8f acc0 = {};
    v8f acc1 = {};
    v2f bones = {1.0f, 1.0f};
    const int mrow = lane & 15;
    const int koff = (lane >> 4) * 2;
#else
    float sfb = 0.0f;
#endif

    for (int k0 = 0; k0 < M_; k0 += 64) {
#ifdef HAVE_ASYNC
        if (k0 + 64 <= M_) {
            // async staging: one b64 per (row, lane); 32 lanes cover 64 floats
            for (int r = 0; r < 16; ++r) {
                const int row = row0 + r;
                if (row < N_) {
                    async_g2l_b64(&Cb[(size_t)row * M_ + k0 + lane * 2],
                                  &tile[r * 68 + lane * 2]);
                } else {
                    tile[r * 68 + lane * 2] = 0.0f;
                    tile[r * 68 + lane * 2 + 1] = 0.0f;
                }
            }
            __builtin_amdgcn_s_wait_asynccnt(0);
        } else
#endif
        {
            // guarded synchronous staging (tail chunk / fallback)
            for (int r = 0; r < 16; ++r) {
                const int row = row0 + r;
                float x0 = 0.0f, x1 = 0.0f;
                const int c0 = k0 + lane, c1 = c0 + 32;
                if (row < N_) {
                    if (c0 < M_) x0 = Cb[(size_t)row * M_ + c0];
                    if (c1 < M_) x1 = Cb[(size_t)row * M_ + c1];
                }
                tile[r * 68 + lane] = x0;
                tile[r * 68 + lane + 32] = x1;
            }
        }
        __syncthreads();
#ifdef HAVE_WMMA4
#pragma unroll
        for (int kk = 0; kk < 64; kk += 8) {
            v2f a0 = *(const v2f*)&tile[mrow * 68 + kk + koff];
            a0.x *= a0.x; a0.y *= a0.y;
            acc0 = __builtin_amdgcn_wmma_f32_16x16x4_f32(
                false, a0, false, bones, (short)0, acc0, false, false);
            v2f a1 = *(const v2f*)&tile[mrow * 68 + kk + 4 + koff];
            a1.x *= a1.x; a1.y *= a1.y;
            acc1 = __builtin_amdgcn_wmma_f32_16x16x4_f32(
                false, a1, false, bones, (short)0, acc1, false, false);
        }
#else
        if (lane < 16) {
            for (int kk = 0; kk < 64; ++kk) {
                float x = tile[lane * 68 + kk];
                sfb = fmaf(x, x, sfb);
            }
        }
#endif
        __syncthreads();
    }

#ifdef HAVE_WMMA4
    v8f acc;
#pragma unroll
    for (int r = 0; r < 8; ++r) acc[r] = acc0[r] + acc1[r];
    if (lane == 0) {
#pragma unroll
        for (int r = 0; r < 8; ++r) red[r] = acc[r];
    }
    if (lane == 16) {
#pragma unroll
        for (int r = 0; r < 8; ++r) red[8 + r] = acc[r];
    }
    __syncthreads();
    if (lane < 16) {
        const int row = row0 + lane;
        if (row < N_) {
            float nrm = sqrtf(red[lane]);
            rinv[(size_t)b * N_ + row] = 1.0f / fmaxf(nrm, 1e-12f);
        }
    }
#else
    if (lane < 16) {
        const int row = row0 + lane;
        if (row < N_) {
            float nrm = sqrtf(sfb);
            rinv[(size_t)b * N_ + row] = 1.0f / fmaxf(nrm, 1e-12f);
        }
    }
#endif
}

// ---------------------------------------------------------------------------
// K2a: per-row 8 smallest (order invariant under positive row scale).
// One wave per row; per-lane sorted-8, then 8-round wave merge.
// ---------------------------------------------------------------------------
__global__ __launch_bounds__(256) void k_rowtopk(const float* __restrict__ C,
                                                 const float* __restrict__ rinv,
                                                 int* __restrict__ rti,
                                                 float* __restrict__ rtv) {
    const int wave = threadIdx.x >> 5;
    const int lane = threadIdx.x & 31;
    const int rowG = blockIdx.x * 8 + wave;  // 0..B_*N_-1
    const float* Crow = C + (size_t)rowG * M_;

    float v[8];
    int id[8];
#pragma unroll
    for (int t = 0; t < 8; ++t) { v[t] = INFINITY; id[t] = 0x7fffffff; }

    for (int j = lane; j < M_; j += 32) {
        float x = Crow[j];
        if (x < v[7]) {
            v[7] = x; id[7] = j;
#pragma unroll
            for (int t = 7; t > 0; --t) {
                if (v[t] < v[t - 1]) {
                    float tv = v[t]; v[t] = v[t - 1]; v[t - 1] = tv;
                    int ti = id[t]; id[t] = id[t - 1]; id[t - 1] = ti;
                }
            }
        }
    }

    const float rv = rinv[rowG];
    int ptr = 0;
    float outV[8]; int outI[8];
#pragma unroll
    for (int r = 0; r < 8; ++r) {
        float cv = (ptr < 8) ? v[ptr] : INFINITY;
        int ci = (ptr < 8) ? id[ptr] : 0x7fffffff;
        float bv = cv; int bi = ci;
#pragma unroll
        for (int off = 16; off >= 1; off >>= 1) {
            float ov = __shfl_xor(bv, off, 32);
            int oi = __shfl_xor(bi, off, 32);
            if (ov < bv || (ov == bv && oi < bi)) { bv = ov; bi = oi; }
        }
        outV[r] = bv; outI[r] = bi;
        if (ptr < 8 && ci == bi) ptr++;  // column index is unique per lane
    }
    if (lane == 0) {
#pragma unroll
        for (int r = 0; r < 8; ++r) {
            rti[(size_t)rowG * 8 + r] = outI[r];
            rtv[(size_t)rowG * 8 + r] = outV[r] * rv;  // store Cn value
        }
    }
}

// ---------------------------------------------------------------------------
// K2b: per-column 8 largest of Cn. Thread per column, row-major streaming.
// ---------------------------------------------------------------------------
__global__ __launch_bounds__(256) void k_coltopk(const float* __restrict__ C,
                                                 const float* __restrict__ rinv,
                                                 int* __restrict__ cti,
                                                 float* __restrict__ ctv) {
    constexpr int BPB = (M_ + 255) / 256;  // 12
    const int b = blockIdx.x / BPB;
    const int j = (blockIdx.x % BPB) * 256 + threadIdx.x;
    if (j >= M_) return;
    const float* Cb = C + (size_t)b * N_ * M_;
    const float* rb = rinv + (size_t)b * N_;

    float v[8];
    int id[8];
#pragma unroll
    for (int t = 0; t < 8; ++t) { v[t] = -INFINITY; id[t] = 0x7fffffff; }

    for (int i = 0; i < N_; ++i) {
        if (i + 8 < N_) __builtin_prefetch(&Cb[(size_t)(i + 8) * M_ + j], 0, 1);
        float x = Cb[(size_t)i * M_ + j] * rb[i];
        if (x > v[7]) {
            v[7] = x; id[7] = i;
#pragma unroll
            for (int t = 7; t > 0; --t) {
                if (v[t] > v[t - 1]) {
                    float tv = v[t]; v[t] = v[t - 1]; v[t - 1] = tv;
                    int ti = id[t]; id[t] = id[t - 1]; id[t - 1] = ti;
                }
            }
        }
    }
#pragma unroll
    for (int r = 0; r < 8; ++r) {
        cti[(size_t)(b * M_ + j) * 8 + r] = id[r];
        ctv[(size_t)(b * M_ + j) * 8 + r] = v[r];
    }
}

// ---------------------------------------------------------------------------
// K3a: init CSR/CSC slots 0..7 from the masks; precompute log(mu/nu + 1e-8).
// ---------------------------------------------------------------------------
__global__ void k_build_init(const int* __restrict__ rti, const float* __restrict__ rtv,
                             const int* __restrict__ cti, const float* __restrict__ ctv,
                             const float* __restrict__ mu, const float* __restrict__ nu,
                             int* rowCnt, int* rowCols, float* rowVals,
                             int* colCnt, int* colRows, float* colVals,
                             float* logmu, float* lognu) {
    const int t = blockIdx.x * blockDim.x + threadIdx.x;
    if (t >= B_ * N_) return;
    rowCnt[t] = 8;
    colCnt[t] = 8;
#pragma unroll
    for (int r = 0; r < 8; ++r) {
        rowCols[(size_t)t * CAP + r] = rti[(size_t)t * 8 + r];
        rowVals[(size_t)t * CAP + r] = rtv[(size_t)t * 8 + r];
        colRows[(size_t)t * CAP + r] = cti[(size_t)t * 8 + r];
        colVals[(size_t)t * CAP + r] = ctv[(size_t)t * 8 + r];
    }
    logmu[t] = logf(mu[t] + 1e-8f);
    lognu[t] = logf(nu[t] + 1e-8f);
}

// K3b: append column-mask entries (i,j) to row i's CSR if j not in rowTop(i).
__global__ void k_append_rows(const int* __restrict__ rti, const int* __restrict__ cti,
                              const float* __restrict__ ctv,
                              int* rowCnt, int* rowCols, float* rowVals) {
    const int t = blockIdx.x * blockDim.x + threadIdx.x;
    if (t >= B_ * M_ * KNN) return;
    const int b = t / (M_ * KNN);
    const int rem = t - b * (M_ * KNN);
    const int j = rem / KNN;
    const int i = cti[t];
    const float val = ctv[t];
    const int* rt = rti + (size_t)(b * N_ + i) * KNN;
    bool dup = false;
#pragma unroll
    for (int q = 0; q < KNN; ++q) dup = dup || (rt[q] == j);
    if (!dup) {
        int pos = atomicAdd(&rowCnt[b * N_ + i], 1);
        if (pos < CAP) {
            rowCols[(size_t)(b * N_ + i) * CAP + pos] = j;
            rowVals[(size_t)(b * N_ + i) * CAP + pos] = val;
        }
    }
}

// K3c: append row-mask entries (i,j) to col j's CSC if i not in colTop(j).
__global__ void k_append_cols(const int* __restrict__ cti, const int* __restrict__ rti,
                              const float* __restrict__ rtv,
                              int* colCnt, int* colRows, float* colVals) {
    const int t = blockIdx.x * blockDim.x + threadIdx.x;
    if (t >= B_ * N_ * KNN) return;
    const int b = t / (N_ * KNN);
    const int rem = t - b * (N_ * KNN);
    const int i = rem / KNN;
    const int j = rti[t];
    const float val = rtv[t];
    const int* ct = cti + (size_t)(b * M_ + j) * KNN;
    bool dup = false;
#pragma unroll
    for (int q = 0; q < KNN; ++q) dup = dup || (ct[q] == i);
    if (!dup) {
        int pos = atomicAdd(&colCnt[b * M_ + j], 1);
        if (pos < CAP) {
            colRows[(size_t)(b * M_ + j) * CAP + pos] = i;
            colVals[(size_t)(b * M_ + j) * CAP + pos] = val;
        }
    }
}

// K3d: clamp counts; sort extra slots by index -> deterministic sum order.
__global__ void k_sortlists(int* rowCnt, int* rowCols, float* rowVals,
                            int* colCnt, int* colRows, float* colVals) {
    const int t = blockIdx.x * blockDim.x + threadIdx.x;
    if (t >= B_ * N_) return;
    {
        int c = rowCnt[t]; if (c > CAP) c = CAP; rowCnt[t] = c;
        const size_t base = (size_t)t * CAP;
        for (int a = 9; a < c; ++a) {
            int key = rowCols[base + a]; float kv = rowVals[base + a];
            int p = a - 1;
            while (p >= 8 && rowCols[base + p] > key) {
                rowCols[base + p + 1] = rowCols[base + p];
                rowVals[base + p + 1] = rowVals[base + p];
                --p;
            }
            rowCols[base + p + 1] = key; rowVals[base + p + 1] = kv;
        }
    }
    {
        int c = colCnt[t]; if (c > CAP) c = CAP; colCnt[t] = c;
        const size_t base = (size_t)t * CAP;
        for (int a = 9; a < c; ++a) {
            int key = colRows[base + a]; float kv = colVals[base + a];
            int p = a - 1;
            while (p >= 8 && colRows[base + p] > key) {
                colRows[base + p + 1] = colRows[base + p];
                colVals[base + p + 1] = colVals[base + p];
                --p;
            }
            colRows[base + p + 1] = key; colVals[base + p + 1] = kv;
        }
    }
}

// ---------------------------------------------------------------------------
// K4: sparse log-domain Sinkhorn, one workgroup per batch, u/v in LDS.
// ---------------------------------------------------------------------------
__global__ __launch_bounds__(1024) void k_sinkhorn(
    const int* __restrict__ rowCnt, const int* __restrict__ rowCols,
    const float* __restrict__ rowVals,
    const int* __restrict__ colCnt, const int* __restrict__ colRows,
    const float* __restrict__ colVals,
    const float* __restrict__ logmu, const float* __restrict__ lognu,
    float* __restrict__ costs) {
    __shared__ float u_s[N_];
    __shared__ float v_s[M_];
    __shared__ float sum_s[N_];
    __shared__ float red_s[1024];

    const int b = blockIdx.x;
    const int tid = threadIdx.x;
    const int base = b * N_;

    for (int i = tid; i < N_; i += 1024) { u_s[i] = 0.0f; v_s[i] = 0.0f; }
    __syncthreads();

    for (int it = 0; it < MAX_ITER; ++it) {
        // row sums with current (u, v)
        for (int i = tid; i < N_; i += 1024) {
            const int c = rowCnt[base + i];
            const int* cols = rowCols + (size_t)(base + i) * CAP;
            const float* vals = rowVals + (size_t)(base + i) * CAP;
            const float ui = u_s[i];
            float s = 0.0f;
            for (int e = 0; e < c; ++e)
                s += expf((ui + v_s[cols[e]] - vals[e]) * INV_EPS);
            sum_s[i] = s;
        }
        __syncthreads();
        // u update + per-thread error partial
        float errp = 0.0f;
        for (int i = tid; i < N_; i += 1024) {
            float un = EPSV * (logmu[base + i] - logf(1e-8f + sum_s[i])) + u_s[i];
            errp += fabsf(un - u_s[i]);
            u_s[i] = un;
        }
        __syncthreads();
        // column sums with u_new; v update
        for (int j = tid; j < M_; j += 1024) {
            const int c = colCnt[base + j];
            const int* rows = colRows + (size_t)(base + j) * CAP;
            const float* vals = colVals + (size_t)(base + j) * CAP;
            const float vj = v_s[j];
            float s = 0.0f;
            for (int e = 0; e < c; ++e)
                s += expf((u_s[rows[e]] + vj - vals[e]) * INV_EPS);
            v_s[j] = EPSV * (lognu[base + j] - logf(1e-8f + s)) + vj;
        }
        // error reduction + uniform early break (matches reference freeze)
        red_s[tid] = errp;
        __syncthreads();
        for (int sft = 512; sft > 0; sft >>= 1) {
            if (tid < sft) red_s[tid] += red_s[tid + sft];
            __syncthreads();
        }
        const float err = red_s[0];
        __syncthreads();
        if (err < THRESH) break;
    }

    // cost = sum over mask entries of val * exp((u+v-val)/eps)
    float cp = 0.0f;
    for (int i = tid; i < N_; i += 1024) {
        const int c = rowCnt[base + i];
        const int* cols = rowCols + (size_t)(base + i) * CAP;
        const float* vals = rowVals + (size_t)(base + i) * CAP;
        const float ui = u_s[i];
        for (int e = 0; e < c; ++e) {
            const float val = vals[e];
            cp += val * expf((ui + v_s[cols[e]] - val) * INV_EPS);
        }
    }
    red_s[tid] = cp;
    __syncthreads();
    for (int sft = 512; sft > 0; sft >>= 1) {
        if (tid < sft) red_s[tid] += red_s[tid + sft];
        __syncthreads();
    }
    if (tid == 0) costs[b] = red_s[0];
}

__global__ void k_final(const float* __restrict__ costs, float* __restrict__ out) {
    if (blockIdx.x == 0 && threadIdx.x == 0) {
        float s = 0.0f;
        for (int b = 0; b < B_; ++b) s += costs[b];
        out[0] = s / (float)B_;
    }
}

// ---------------------------------------------------------------------------
extern "C" void kernel_launch(void* const* d_in, const int* in_sizes, int n_in,
                              void* d_out, int out_size, void* d_ws, size_t ws_size,
                              hipStream_t stream) {
    (void)in_sizes; (void)n_in; (void)out_size; (void)ws_size;
    const float* mu = (const float*)d_in[0];
    const float* nu = (const float*)d_in[1];
    const float* C = (const float*)d_in[2];
    float* out = (float*)d_out;

    char* p = (char*)d_ws;
    auto take = [&](size_t nElem) { void* r = (void*)p; p += nElem * 4; return r; };
    float* rinv   = (float*)take((size_t)B_ * N_);
    int*   rti    = (int*)take((size_t)B_ * N_ * KNN);
    float* rtv    = (float*)take((size_t)B_ * N_ * KNN);
    int*   cti    = (int*)take((size_t)B_ * M_ * KNN);
    float* ctv    = (float*)take((size_t)B_ * M_ * KNN);
    int*   rowCnt = (int*)take((size_t)B_ * N_);
    int*   colCnt = (int*)take((size_t)B_ * M_);
    int*   rowCols = (int*)take((size_t)B_ * N_ * CAP);
    float* rowVals = (float*)take((size_t)B_ * N_ * CAP);
    int*   colRows = (int*)take((size_t)B_ * M_ * CAP);
    float* colVals = (float*)take((size_t)B_ * M_ * CAP);
    float* logmu  = (float*)take((size_t)B_ * N_);
    float* lognu  = (float*)take((size_t)B_ * M_);
    float* costs  = (float*)take((size_t)B_);

    k_rownorm<<<B_ * ((N_ + 15) / 16), 32, 0, stream>>>(C, rinv);
    k_rowtopk<<<(B_ * N_) / 8, 256, 0, stream>>>(C, rinv, rti, rtv);
    k_coltopk<<<B_ * ((M_ + 255) / 256), 256, 0, stream>>>(C, rinv, cti, ctv);
    k_build_init<<<(B_ * N_ + 255) / 256, 256, 0, stream>>>(
        rti, rtv, cti, ctv, mu, nu, rowCnt, rowCols, rowVals, colCnt, colRows,
        colVals, logmu, lognu);
    k_append_rows<<<(B_ * M_ * KNN + 255) / 256, 256, 0, stream>>>(
        rti, cti, ctv, rowCnt, rowCols, rowVals);
    k_append_cols<<<(B_ * N_ * KNN + 255) / 256, 256, 0, stream>>>(
        cti, rti, rtv, colCnt, colRows, colVals);
    k_sortlists<<<(B_ * N_ + 255) / 256, 256, 0, stream>>>(
        rowCnt, rowCols, rowVals, colCnt, colRows, colVals);
    k_sinkhorn<<<B_, 1024, 0, stream>>>(rowCnt, rowCols, rowVals, colCnt,
                                        colRows, colVals, logmu, lognu, costs);
    k_final<<<1, 32, 0, stream>>>(costs, out);
}